// B_GraphAttentionLayer_43662637531197
// MI455X (gfx1250) — compile-verified
//
#include <hip/hip_runtime.h>
#include <math.h>

#define B 8
#define N 2048
#define NW (N / 32)          // packed words per adjacency row
#define NSPLIT 8             // n-loop split factor for the adj streaming pass
#define NSEG (N / NSPLIT)    // rows per segment
#define FIN 256
#define FOUT 64
#define ALPHA 0.2f
#define NEG_INF_F (-9.0e15f)

typedef __attribute__((ext_vector_type(2))) float v2f;
typedef __attribute__((ext_vector_type(8))) float v8f;

// ---------------------------------------------------------------------------
// Kernel 1: Wh[b] = h[b] @ W[b]    ([N,FIN] @ [FIN,FOUT] -> [N,FOUT])
// f32 WMMA 16x16x4. One wave per 16-row tile; 4 accumulators cover FOUT=64.
// ---------------------------------------------------------------------------
__global__ __launch_bounds__(128) void wh_gemm_kernel(const float* __restrict__ h,
                                                      const float* __restrict__ Wmat,
                                                      float* __restrict__ Wh) {
  const int wave = threadIdx.x >> 5;
  const int lane = threadIdx.x & 31;
  const int half = lane >> 4;           // K-half selector (A/B frags)
  const int l    = lane & 15;
  const int tileGlobal = blockIdx.x * 4 + wave;   // 0 .. B*N/16-1
  const int b    = tileGlobal >> 7;               // 128 tiles per batch
  const int tile = tileGlobal & 127;
  const int row  = tile * 16 + l;

  const float* __restrict__ hrow = h + ((size_t)b * N + row) * FIN;
  const float* __restrict__ Wb   = Wmat + (size_t)b * FIN * FOUT;

  v8f acc0 = {}, acc1 = {}, acc2 = {}, acc3 = {};
  for (int kk = 0; kk < FIN; kk += 4) {
    const int k0 = kk + 2 * half;
    // A fragment: 16x4 f32 — lane row = l, K = k0,k0+1 (contiguous -> b64)
    v2f a = *(const v2f*)(hrow + k0);
    // B fragments: 4x16 f32 — lane col = l(+16*ct), K = k0,k0+1 across v0/v1
    const float* w0  = Wb + (size_t)k0 * FOUT + l;
    const float* w1p = w0 + FOUT;
    v2f b0; b0.x = w0[0];  b0.y = w1p[0];
    v2f b1; b1.x = w0[16]; b1.y = w1p[16];
    v2f b2; b2.x = w0[32]; b2.y = w1p[32];
    v2f b3; b3.x = w0[48]; b3.y = w1p[48];
    acc0 = __builtin_amdgcn_wmma_f32_16x16x4_f32(false, a, false, b0, (short)0, acc0, false, false);
    acc1 = __builtin_amdgcn_wmma_f32_16x16x4_f32(false, a, false, b1, (short)0, acc1, false, false);
    acc2 = __builtin_amdgcn_wmma_f32_16x16x4_f32(false, a, false, b2, (short)0, acc2, false, false);
    acc3 = __builtin_amdgcn_wmma_f32_16x16x4_f32(false, a, false, b3, (short)0, acc3, false, false);
  }
  // D layout: VGPR v -> M = v + 8*half, N = l (+16*ct)
  float* outBase = Wh + ((size_t)b * N + (size_t)tile * 16) * FOUT;
  #pragma unroll
  for (int v = 0; v < 8; ++v) {
    float* orow = outBase + (size_t)(v + 8 * half) * FOUT + l;
    orow[0]  = acc0[v];
    orow[16] = acc1[v];
    orow[32] = acc2[v];
    orow[48] = acc3[v];
  }
}

// ---------------------------------------------------------------------------
// Kernel 2: Wh1[b,n] = Wh[b,n,:] . a[b,:64] ; Wh2[b,n] = Wh[b,n,:] . a[b,64:]
// ---------------------------------------------------------------------------
__global__ __launch_bounds__(256) void attn_vec_kernel(const float* __restrict__ Wh,
                                                       const float* __restrict__ avec,
                                                       float* __restrict__ w1,
                                                       float* __restrict__ w2) {
  const int idx = blockIdx.x * blockDim.x + threadIdx.x;  // b*N + n
  if (idx >= B * N) return;
  const int b = idx >> 11;  // / N
  const float* __restrict__ whr = Wh + (size_t)idx * FOUT;
  const float* __restrict__ ab  = avec + (size_t)b * 2 * FOUT;
  float s1 = 0.f, s2 = 0.f;
  #pragma unroll 8
  for (int o = 0; o < FOUT; ++o) {
    const float v = whr[o];
    s1 += v * ab[o];
    s2 += v * ab[FOUT + o];
  }
  w1[idx] = s1;
  w2[idx] = s2;
}

// ---------------------------------------------------------------------------
// Kernel 3: THE single pass over adj (134MB). Split 8-ways in n for wave-level
// concurrency (4096 waves, each 256 fully-coalesced 128B wave-loads):
//  - per-(column, segment) online softmax partials (softmax axis=1)
//  - wave32 ballot packs adj into a bitmask (32x compression, L2-resident)
// Lane <-> column m; each wave is 32 consecutive m within one n-segment, so
// the ballot word is exactly abits[b, n, m>>5] and segments are disjoint in n.
// ---------------------------------------------------------------------------
__global__ __launch_bounds__(256) void col_stats_pack_kernel(const int* __restrict__ adj,
                                                             const float* __restrict__ w1,
                                                             const float* __restrict__ w2,
                                                             float* __restrict__ pmax,
                                                             float* __restrict__ psum,
                                                             unsigned* __restrict__ abits) {
  const int tid  = blockIdx.x * blockDim.x + threadIdx.x;  // (b*NSPLIT+seg)*N + m
  const int m    = tid & (N - 1);
  const int seg  = (tid >> 11) & (NSPLIT - 1);
  const int b    = tid >> 14;                              // / (NSPLIT*N)
  const int lane = threadIdx.x & 31;
  const int n0   = seg * NSEG;
  const int* __restrict__ acol = adj + (size_t)b * N * N + (size_t)n0 * N + m;  // stride N
  unsigned* __restrict__ awcol = abits + (size_t)b * N * NW + (size_t)n0 * NW + (m >> 5);
  const float w2m = w2[b * N + m];
  const float* __restrict__ w1b = w1 + (size_t)b * N + n0;
  float M = NEG_INF_F;
  float S = 0.f;
  #pragma unroll 4
  for (int n = 0; n < NSEG; ++n) {
    const int av = acol[(size_t)n * N];
    const unsigned ball = __builtin_amdgcn_ballot_w32(av > 0);
    if (lane == 0) awcol[(size_t)n * NW] = ball;
    if (av > 0) {
      float x = w1b[n] + w2m;
      x = x > 0.f ? x : ALPHA * x;
      if (x > M) {
        S = S * __expf(M - x) + 1.f;   // first hit: exp(-huge) == 0 -> S = 1
        M = x;
      } else {
        S += __expf(x - M);
      }
    }
  }
  pmax[tid] = M;   // tid == (b*NSPLIT+seg)*N + m, the natural partial index
  psum[tid] = S;
}

// ---------------------------------------------------------------------------
// Kernel 3b: merge NSPLIT online-softmax partials per column.
// M = max_i M_i ; S = sum_i S_i * exp(M_i - M)   (M_i - M <= 0 always)
// ---------------------------------------------------------------------------
__global__ __launch_bounds__(256) void merge_stats_kernel(const float* __restrict__ pmax,
                                                          const float* __restrict__ psum,
                                                          float* __restrict__ colmax,
                                                          float* __restrict__ colsum) {
  const int idx = blockIdx.x * blockDim.x + threadIdx.x;  // b*N + m
  if (idx >= B * N) return;
  const int b = idx >> 11;
  const int m = idx & (N - 1);
  const float* __restrict__ pm = pmax + (size_t)b * NSPLIT * N + m;
  const float* __restrict__ ps = psum + (size_t)b * NSPLIT * N + m;
  float M = NEG_INF_F;
  #pragma unroll
  for (int i = 0; i < NSPLIT; ++i) M = fmaxf(M, pm[(size_t)i * N]);
  float S = 0.f;
  #pragma unroll
  for (int i = 0; i < NSPLIT; ++i) S += ps[(size_t)i * N] * __expf(pm[(size_t)i * N] - M);
  colmax[idx] = M;
  colsum[idx] = (S > 1e-30f) ? S : 1.f;   // guard the (practically impossible) empty column
}

// ---------------------------------------------------------------------------
// Kernel 4: WhsT[b,o,m] = Wh[b,m,o] / colsum[b,m]
// Transposed + pre-scaled so GEMM-2 B-fragments are contiguous K-pairs (b64).
// ---------------------------------------------------------------------------
__global__ __launch_bounds__(256) void scale_whT_kernel(const float* __restrict__ Wh,
                                                        const float* __restrict__ colsum,
                                                        float* __restrict__ WhsT) {
  const int idx = blockIdx.x * blockDim.x + threadIdx.x;  // b*FOUT*N + o*N + m
  if (idx >= B * FOUT * N) return;
  const int m  = idx & (N - 1);
  const int o  = (idx >> 11) & (FOUT - 1);
  const int b  = idx >> 17;                // / (FOUT*N)
  WhsT[idx] = Wh[((size_t)b * N + m) * FOUT + o] * (1.0f / colsum[b * N + m]);
}

// ---------------------------------------------------------------------------
// Kernel 5: h_prime[b,n,:] = sum_m s[n,m] * Whs[m,:]; s built on the fly from
// packed adjacency bits + attention scalars + colmax; fused ELU epilogue.
// f32 WMMA 16x16x4; one packed word covers 8 K-steps per lane.
// ---------------------------------------------------------------------------
__global__ __launch_bounds__(128) void attn_gemm_kernel(const unsigned* __restrict__ abits,
                                                        const float* __restrict__ w1,
                                                        const float* __restrict__ w2,
                                                        const float* __restrict__ colmax,
                                                        const float* __restrict__ WhsT,
                                                        float* __restrict__ out) {
  const int wave = threadIdx.x >> 5;
  const int lane = threadIdx.x & 31;
  const int half = lane >> 4;
  const int l    = lane & 15;
  const int tileGlobal = blockIdx.x * 4 + wave;
  const int b    = tileGlobal >> 7;
  const int tile = tileGlobal & 127;
  const int row  = tile * 16 + l;

  const unsigned* __restrict__ brow = abits + ((size_t)b * N + row) * NW;
  const float  w1r = w1[(size_t)b * N + row];
  const float* __restrict__ w2b = w2 + (size_t)b * N;
  const float* __restrict__ cmb = colmax + (size_t)b * N;
  const float* __restrict__ Tb  = WhsT + (size_t)b * FOUT * N;
  // Per-lane B-fragment row pointers (column of D = row of WhsT), advance in m
  const float* __restrict__ t0 = Tb + (size_t)(l)      * N;
  const float* __restrict__ t1 = Tb + (size_t)(l + 16) * N;
  const float* __restrict__ t2 = Tb + (size_t)(l + 32) * N;
  const float* __restrict__ t3 = Tb + (size_t)(l + 48) * N;

  v8f acc0 = {}, acc1 = {}, acc2 = {}, acc3 = {};
  for (int chunk = 0; chunk < NW; ++chunk) {
    const unsigned bits = brow[chunk];          // adjacency bits for 32 columns
    const int mbase = chunk * 32;
    __builtin_prefetch(t0 + mbase + 512, 0, 1); // stream WhsT ahead (L2-hot)
    #pragma unroll
    for (int j = 0; j < 8; ++j) {
      const int c0   = mbase + j * 4 + 2 * half;
      const int bit0 = j * 4 + 2 * half;
      // A fragment: masked, shifted, exponentiated scores (branch-free)
      const v2f w2p = *(const v2f*)(w2b + c0);
      const v2f cmp = *(const v2f*)(cmb + c0);
      float x0 = w1r + w2p.x; x0 = x0 > 0.f ? x0 : ALPHA * x0;
      float x1 = w1r + w2p.y; x1 = x1 > 0.f ? x1 : ALPHA * x1;
      v2f av;
      av.x = ((bits >> bit0) & 1u)       ? __expf(x0 - cmp.x) : 0.f;
      av.y = ((bits >> (bit0 + 1)) & 1u) ? __expf(x1 - cmp.y) : 0.f;
      // B fragments: contiguous K-pairs from transposed, pre-scaled Wh
      const v2f b0 = *(const v2f*)(t0 + c0);
      const v2f b1 = *(const v2f*)(t1 + c0);
      const v2f b2 = *(const v2f*)(t2 + c0);
      const v2f b3 = *(const v2f*)(t3 + c0);
      acc0 = __builtin_amdgcn_wmma_f32_16x16x4_f32(false, av, false, b0, (short)0, acc0, false, false);
      acc1 = __builtin_amdgcn_wmma_f32_16x16x4_f32(false, av, false, b1, (short)0, acc1, false, false);
      acc2 = __builtin_amdgcn_wmma_f32_16x16x4_f32(false, av, false, b2, (short)0, acc2, false, false);
      acc3 = __builtin_amdgcn_wmma_f32_16x16x4_f32(false, av, false, b3, (short)0, acc3, false, false);
    }
  }
  // ELU epilogue + store
  float* outBase = out + ((size_t)b * N + (size_t)tile * 16) * FOUT;
  #pragma unroll
  for (int v = 0; v < 8; ++v) {
    float* orow = outBase + (size_t)(v + 8 * half) * FOUT + l;
    float e0 = acc0[v]; orow[0]  = e0 > 0.f ? e0 : expm1f(e0);
    float e1 = acc1[v]; orow[16] = e1 > 0.f ? e1 : expm1f(e1);
    float e2 = acc2[v]; orow[32] = e2 > 0.f ? e2 : expm1f(e2);
    float e3 = acc3[v]; orow[48] = e3 > 0.f ? e3 : expm1f(e3);
  }
}

// ---------------------------------------------------------------------------
extern "C" void kernel_launch(void* const* d_in, const int* in_sizes, int n_in,
                              void* d_out, int out_size, void* d_ws, size_t ws_size,
                              hipStream_t stream) {
  (void)in_sizes; (void)n_in; (void)out_size; (void)ws_size;
  const float* h    = (const float*)d_in[0];   // [B,N,FIN]
  const int*   adj  = (const int*)d_in[1];     // [B,N,N]
  const float* Wmat = (const float*)d_in[2];   // [B,FIN,FOUT]
  const float* avec = (const float*)d_in[3];   // [B,2*FOUT,1]
  float* out = (float*)d_out;                  // [B,N,FOUT]

  float* ws    = (float*)d_ws;
  float* Wh    = ws;                           // B*N*FOUT
  float* WhsT  = Wh   + (size_t)B * N * FOUT;  // B*FOUT*N (transposed, scaled)
  float* w1    = WhsT + (size_t)B * FOUT * N;  // B*N
  float* w2    = w1   + (size_t)B * N;         // B*N
  float* cmax  = w2   + (size_t)B * N;         // B*N
  float* csum  = cmax + (size_t)B * N;         // B*N
  float* pmax  = csum + (size_t)B * N;         // B*NSPLIT*N
  float* psum  = pmax + (size_t)B * NSPLIT * N;// B*NSPLIT*N
  unsigned* abits = (unsigned*)(psum + (size_t)B * NSPLIT * N);  // B*N*NW words

  const int nTiles = B * (N / 16);             // 1024 wave-tiles, 4 per block
  wh_gemm_kernel       <<<nTiles / 4, 128, 0, stream>>>(h, Wmat, Wh);
  attn_vec_kernel      <<<(B * N + 255) / 256, 256, 0, stream>>>(Wh, avec, w1, w2);
  col_stats_pack_kernel<<<(B * NSPLIT * N) / 256, 256, 0, stream>>>(adj, w1, w2, pmax, psum, abits);
  merge_stats_kernel   <<<(B * N + 255) / 256, 256, 0, stream>>>(pmax, psum, cmax, csum);
  scale_whT_kernel     <<<(B * FOUT * N + 255) / 256, 256, 0, stream>>>(Wh, csum, WhsT);
  attn_gemm_kernel     <<<nTiles / 4, 128, 0, stream>>>(abits, w1, w2, cmax, WhsT, out);
}